// GCU_18202071401057
// MI455X (gfx1250) — compile-verified
//
#include <hip/hip_runtime.h>
#include <math.h>

// Problem constants (match reference)
#define HW    16384   // H*Wd = 128*128
#define DD    64      // D
#define VV    64      // V
#define OUTF  32
#define EPSV  1e-7f

typedef float v2f __attribute__((ext_vector_type(2)));
typedef float v8f __attribute__((ext_vector_type(8)));

// ---- workspace layout (float offsets) ----
#define WS_Q      0            // 16384*64   = 1048576
#define WS_COEF   1048576      // 3*64       (A, B, C)
#define WS_U      1048768      // 64*64
#define WS_T      1052864      // 64*64*64   = 262144   T[(v*64+d')*64+d]
#define WS_Z      1315008      // 64*64      Z[d*64+v]
#define WS_OUT1   1319104      // 64*32
#define WS_M      1321152      // 64*32
#define WS_ZO     1323200      // 64*32

// -------------------------------------------------------------------------
// K1: per-v quadratic coefficients  A[v]=sum_d iv^2, B[v]=sum_d W iv^2,
//     C[v]=sum_d W^2 iv^2    (1 block, 64 threads)
// -------------------------------------------------------------------------
__global__ void coef_kernel(const float* __restrict__ W,
                            const float* __restrict__ var,
                            float* __restrict__ coef) {
  int v = threadIdx.x;
  float A = 0.f, Bc = 0.f, Cc = 0.f;
  for (int d = 0; d < DD; ++d) {
    float w  = W[d * VV + v];
    float iv = 1.0f / var[d * VV + v];
    float iv2 = iv * iv;
    A  += iv2;
    Bc += w * iv2;
    Cc += w * w * iv2;
  }
  coef[v]       = A;
  coef[64 + v]  = Bc;
  coef[128 + v] = Cc;
}

// -------------------------------------------------------------------------
// K2: soft assignment Q[p,v].  One wave32 per pixel; lane handles v=lane and
//     v=lane+32.  Row min / row sum via wave32 xor butterflies.
// -------------------------------------------------------------------------
__global__ void q_kernel(const float* __restrict__ X,
                         const float* __restrict__ coef,
                         float* __restrict__ Q) {
  int wave = threadIdx.x >> 5;
  int lane = threadIdx.x & 31;
  int p = blockIdx.x * 8 + wave;   // 2048 blocks * 8 waves = 16384 pixels

  int v0 = lane, v1 = lane + 32;
  float x0 = X[p * DD + v0];
  float x1 = X[p * DD + v1];
  float q0 = coef[v0] * x0 * x0 - 2.0f * coef[64 + v0] * x0 + coef[128 + v0];
  float q1 = coef[v1] * x1 * x1 - 2.0f * coef[64 + v1] * x1 + coef[128 + v1];

  float m = fminf(q0, q1);
  #pragma unroll
  for (int mask = 1; mask < 32; mask <<= 1)
    m = fminf(m, __shfl_xor(m, mask, 32));

  float e0 = __expf(-0.5f * (q0 - m));
  float e1 = __expf(-0.5f * (q1 - m));
  float s = e0 + e1;
  #pragma unroll
  for (int mask = 1; mask < 32; mask <<= 1)
    s += __shfl_xor(s, mask, 32);

  float inv = 1.0f / s;
  Q[p * VV + v0] = e0 * inv;
  Q[p * VV + v1] = e1 * inv;
}

// -------------------------------------------------------------------------
// K3: U[d,v] = sum_a Q[(a*64+d), v]   (grid 64 = d, block 64 = v)
// -------------------------------------------------------------------------
__global__ void u_kernel(const float* __restrict__ Q, float* __restrict__ U) {
  int d = blockIdx.x, v = threadIdx.x;
  float acc = 0.f;
  for (int a = 0; a < 256; ++a)
    acc += Q[(a * 64 + d) * VV + v];
  U[d * VV + v] = acc;
}

// -------------------------------------------------------------------------
// K4: 64 independent f32 GEMMs  T_v (64x64) = X_v (64x256) @ Q_v (256x64)
//     via V_WMMA_F32_16X16X4_F32.  One block per v; 16 waves, one 16x16
//     output tile each; K loop of 64 steps of 4.
//     X_v[d',a] = X[(d'*256+a)*64 + v];  Q_v[a,d] = Q[(a*64+d)*64 + v]
// -------------------------------------------------------------------------
__global__ void t_gemm_kernel(const float* __restrict__ X,
                              const float* __restrict__ Q,
                              float* __restrict__ T) {
  int v    = blockIdx.x;
  int wave = threadIdx.x >> 5;
  int lane = threadIdx.x & 31;
  int mT = wave >> 2;          // 4x4 tile grid over 64x64 output
  int nT = wave & 3;
  int half = lane >> 4;        // K half select per ISA A/B layout
  int lr   = lane & 15;

  v8f c = {};
  for (int kk = 0; kk < 64; ++kk) {
    int k0 = kk * 4;
    // A fragment: 16x4 of X_v. lane<16 -> K=k0,k0+1 ; lane>=16 -> K=k0+2,k0+3
    int aRow = mT * 16 + lr;
    int ka   = k0 + half * 2;
    v2f a, b;
    a.x = X[(aRow * 256 + ka    ) * DD + v];
    a.y = X[(aRow * 256 + ka + 1) * DD + v];
    // B fragment: 4x16 of Q_v. vgpr0: K=k0 (lo half) / k0+2 (hi half); vgpr1: +1
    int bCol = nT * 16 + lr;
    int kb   = k0 + half * 2;
    b.x = Q[((kb    ) * 64 + bCol) * VV + v];
    b.y = Q[((kb + 1) * 64 + bCol) * VV + v];
    c = __builtin_amdgcn_wmma_f32_16x16x4_f32(false, a, false, b,
                                              (short)0, c, false, false);
  }
  // C/D layout: VGPR i -> row M=i (lanes 0-15) or M=i+8 (lanes 16-31), col N=lr
  #pragma unroll
  for (int i = 0; i < 8; ++i) {
    int dp = mT * 16 + half * 8 + i;
    int dd = nT * 16 + lr;
    T[(v * 64 + dp) * 64 + dd] = c[i];
  }
}

// -------------------------------------------------------------------------
// K5: z[d',v] = sum_d iv*(T_v[d',d] - W*U) ; normalize by (sumQ+eps); then
//     Z = z / (sum_d' z^2).   grid 64 = v, block 64 = d'
// -------------------------------------------------------------------------
__global__ void z_kernel(const float* __restrict__ T,
                         const float* __restrict__ W,
                         const float* __restrict__ var,
                         const float* __restrict__ U,
                         float* __restrict__ Z) {
  __shared__ float sh[64];
  __shared__ float nrm;
  int v = blockIdx.x, dp = threadIdx.x;

  float sumQ = 0.f;
  for (int d = 0; d < DD; ++d) sumQ += U[d * VV + v];

  float acc = 0.f;
  for (int d = 0; d < DD; ++d) {
    float iv = 1.0f / var[d * VV + v];
    acc += iv * (T[(v * 64 + dp) * 64 + d] - W[d * VV + v] * U[d * VV + v]);
  }
  float z = (acc + EPSV) / (sumQ + EPSV);
  sh[dp] = z * z;
  __syncthreads();
  if (dp == 0) {
    float n = 0.f;
    for (int i = 0; i < 64; ++i) n += sh[i];
    nrm = n;
  }
  __syncthreads();
  Z[dp * VV + v] = z / nrm;
}

// -------------------------------------------------------------------------
// K6a/b/c: tiny tail GEMMs (Adj fused away):
//   out1[v,o] = sum_d Z[d,v]*weight[d,o]
//   M[d,o]    = sum_w Z[d,w]*out1[w,o]         (= Z @ out1)
//   Zo[v,o]   = relu(sum_d Z[d,v]*M[d,o])      (= Adj @ out1, relu)
// each: grid 8, block 256 -> 2048 (row,col) pairs
// -------------------------------------------------------------------------
__global__ void out1_kernel(const float* __restrict__ Z,
                            const float* __restrict__ weight,
                            float* __restrict__ out1) {
  int t = blockIdx.x * 256 + threadIdx.x;
  int v = t >> 5, o = t & 31;
  float acc = 0.f;
  for (int d = 0; d < DD; ++d) acc += Z[d * VV + v] * weight[d * OUTF + o];
  out1[v * OUTF + o] = acc;
}

__global__ void m_kernel(const float* __restrict__ Z,
                         const float* __restrict__ out1,
                         float* __restrict__ M) {
  int t = blockIdx.x * 256 + threadIdx.x;
  int d = t >> 5, o = t & 31;
  float acc = 0.f;
  for (int w = 0; w < VV; ++w) acc += Z[d * VV + w] * out1[w * OUTF + o];
  M[d * OUTF + o] = acc;
}

__global__ void zo_kernel(const float* __restrict__ Z,
                          const float* __restrict__ M,
                          float* __restrict__ Zo) {
  int t = blockIdx.x * 256 + threadIdx.x;
  int v = t >> 5, o = t & 31;
  float acc = 0.f;
  for (int d = 0; d < DD; ++d) acc += Z[d * VV + v] * M[d * OUTF + o];
  Zo[v * OUTF + o] = fmaxf(acc, 0.f);
}

// -------------------------------------------------------------------------
// K7: X_new = Q (16384x64) @ Zo (64x32) via V_WMMA_F32_16X16X4_F32.
//     2048 tiles (1024 M-tiles x 2 N-tiles); 8 waves/block, 256 blocks.
//     Output flat (HW,OUTF) == flat (B,OUTF,H,W) after the raw reshape.
// -------------------------------------------------------------------------
__global__ void xnew_kernel(const float* __restrict__ Q,
                            const float* __restrict__ Zo,
                            float* __restrict__ out) {
  int wave = threadIdx.x >> 5;
  int lane = threadIdx.x & 31;
  int tile = blockIdx.x * 8 + wave;
  int mT = tile >> 1;
  int nT = tile & 1;
  int half = lane >> 4;
  int lr   = lane & 15;

  v8f c = {};
  for (int kk = 0; kk < 16; ++kk) {
    int k0 = kk * 4;
    if (kk + 1 < 16)   // gfx1250 global_prefetch for next A slab
      __builtin_prefetch(&Q[(mT * 16 + lr) * DD + k0 + 4], 0, 1);
    int ka = k0 + half * 2;
    v2f a, b;
    a.x = Q[(mT * 16 + lr) * DD + ka];
    a.y = Q[(mT * 16 + lr) * DD + ka + 1];
    int kb = k0 + half * 2;
    b.x = Zo[(kb    ) * OUTF + nT * 16 + lr];
    b.y = Zo[(kb + 1) * OUTF + nT * 16 + lr];
    c = __builtin_amdgcn_wmma_f32_16x16x4_f32(false, a, false, b,
                                              (short)0, c, false, false);
  }
  #pragma unroll
  for (int i = 0; i < 8; ++i) {
    int row = mT * 16 + half * 8 + i;
    int col = nT * 16 + lr;
    out[row * OUTF + col] = c[i];
  }
}

// -------------------------------------------------------------------------
extern "C" void kernel_launch(void* const* d_in, const int* in_sizes, int n_in,
                              void* d_out, int out_size, void* d_ws, size_t ws_size,
                              hipStream_t stream) {
  const float* X      = (const float*)d_in[0];   // (1,128,128,64)
  const float* W      = (const float*)d_in[1];   // (64,64)
  const float* var    = (const float*)d_in[2];   // (64,64)
  const float* weight = (const float*)d_in[3];   // (64,32)
  float* out = (float*)d_out;                    // (1,32,128,128) flat == (16384,32)
  float* ws  = (float*)d_ws;

  float* Qb   = ws + WS_Q;
  float* coef = ws + WS_COEF;
  float* U    = ws + WS_U;
  float* T    = ws + WS_T;
  float* Z    = ws + WS_Z;
  float* out1 = ws + WS_OUT1;
  float* M    = ws + WS_M;
  float* Zo   = ws + WS_ZO;

  coef_kernel  <<<1,    64,  0, stream>>>(W, var, coef);
  q_kernel     <<<2048, 256, 0, stream>>>(X, coef, Qb);
  u_kernel     <<<64,   64,  0, stream>>>(Qb, U);
  t_gemm_kernel<<<64,   512, 0, stream>>>(X, Qb, T);
  z_kernel     <<<64,   64,  0, stream>>>(T, W, var, U, Z);
  out1_kernel  <<<8,    256, 0, stream>>>(Z, weight, out1);
  m_kernel     <<<8,    256, 0, stream>>>(Z, out1, M);
  zo_kernel    <<<8,    256, 0, stream>>>(Z, M, Zo);
  xnew_kernel  <<<256,  256, 0, stream>>>(Qb, Zo, out);
}